// RelateModule_56367150793557
// MI455X (gfx1250) — compile-verified
//
#include <hip/hip_runtime.h>
#include <hip/hip_bf16.h>
#include <math.h>

// ---------------------------------------------------------------------------
// RelateModule for MI455X (gfx1250, wave32).
//
// Roofline: feat_edge read (536 MB) dominates; everything else is < 3 MB of
// traffic and < 0.5 GFLOP. The big kernel is written as a pure streaming
// GEMV (b128 loads + butterfly reduce). The mask-power batched matmuls
// (128x128x128, f32) use V_WMMA_F32_16X16X4_F32 from LDS.
// ---------------------------------------------------------------------------

#define B_   32
#define NF_  128
#define DE_  256
#define DH_  512
#define G_   2
#define SL_  8

typedef __attribute__((ext_vector_type(2))) float v2f;
typedef __attribute__((ext_vector_type(8))) float v8f;

// ---------------------------------------------------------------------------
// Kernel A: query[b,e] = c_i[b,:] . map_c_w[e,:] + map_c_b[e]   (32x256)
// ---------------------------------------------------------------------------
__global__ __launch_bounds__(256) void query_kernel(
    const float* __restrict__ ci, const float* __restrict__ wmat,
    const float* __restrict__ bias, float* __restrict__ q) {
  int idx = blockIdx.x * 256 + threadIdx.x;       // B_*DE_ = 8192
  int b = idx >> 8;
  int e = idx & 255;
  const float4* c4 = (const float4*)(ci + (size_t)b * DH_);
  const float4* w4 = (const float4*)(wmat + (size_t)e * DH_);
  float s = bias[e];
  for (int h = 0; h < DH_ / 4; ++h) {
    float4 cv = c4[h], wv = w4[h];
    s += cv.x * wv.x + cv.y * wv.y + cv.z * wv.z + cv.w * wv.w;
  }
  q[idx] = s;
}

// ---------------------------------------------------------------------------
// Kernel B: att_in[b,n,g] = sum_s att_stack[b,n,g,s] * stack_ptr[b,s]
// ---------------------------------------------------------------------------
__global__ __launch_bounds__(256) void attin_kernel(
    const float* __restrict__ stk, const float* __restrict__ sp,
    float* __restrict__ attIn) {
  int idx = blockIdx.x * 256 + threadIdx.x;       // B_*NF_*G_ = 8192
  int b = idx >> 8;                               // NF_*G_ = 256 per batch
  const float* s8 = stk + (size_t)idx * SL_;
  const float* p8 = sp + (size_t)b * SL_;
  float s = 0.f;
#pragma unroll
  for (int i = 0; i < SL_; ++i) s += s8[i] * p8[i];
  attIn[idx] = s;
}

// ---------------------------------------------------------------------------
// Kernel C: final_mask = M + 0.9*M^2 + 0.81*M^3 + 0.729*M^4  (per batch)
// One workgroup (8 waves) per batch. M and the running power live in LDS
// (2 x 64 KB). Each wave owns a 16-row strip; WMMA f32 16x16x4, K-loop by 4.
// C/D fragment layout (32-bit, 16x16): lane<16 -> N=lane, VGPR r -> M=r;
// lane>=16 -> N=lane-16, VGPR r -> M=r+8.
// A 16x4 layout: lane -> M=lane&15; v0 holds K=(lane<16?0:2), v1 holds K+1.
// ---------------------------------------------------------------------------
__global__ __launch_bounds__(256) void mask_power_kernel(
    const float* __restrict__ rel, float* __restrict__ finalMask) {
  __shared__ float ldsM[NF_ * NF_];    // 64 KB
  __shared__ float ldsCur[NF_ * NF_];  // 64 KB
  int b = blockIdx.x;
  const float* Mg = rel + (size_t)b * NF_ * NF_;
  for (int i = threadIdx.x; i < NF_ * NF_; i += 256) {
    float v = Mg[i];
    ldsM[i] = v;
    ldsCur[i] = v;                     // cur starts as M
  }
  __syncthreads();

  const int lane  = threadIdx.x & 31;
  const int wave  = threadIdx.x >> 5;
  const int rBase = wave * 16;         // this wave's output row strip
  const int mrow  = lane & 15;
  const int khalf = (lane >> 4) << 1;  // 0 or 2
  const int rofs  = (lane >> 4) * 8;   // D-layout row offset
  const int colIn = lane & 15;

  v8f fin[8];
#pragma unroll
  for (int t = 0; t < 8; ++t) {
    int col = t * 16 + colIn;
#pragma unroll
    for (int r = 0; r < 8; ++r)
      fin[t][r] = ldsM[(rBase + rofs + r) * NF_ + col];
  }

  float decay = 0.9f;
  for (int p = 0; p < 3; ++p) {
    v8f nxt[8];
#pragma unroll
    for (int t = 0; t < 8; ++t) {
      v8f acc = {};
      int nCol = t * 16 + colIn;
      for (int k0 = 0; k0 < NF_; k0 += 4) {
        v2f a, bb;
        a.x  = ldsCur[(rBase + mrow) * NF_ + k0 + khalf];
        a.y  = ldsCur[(rBase + mrow) * NF_ + k0 + khalf + 1];
        bb.x = ldsM[(k0 + khalf) * NF_ + nCol];
        bb.y = ldsM[(k0 + khalf + 1) * NF_ + nCol];
        acc = __builtin_amdgcn_wmma_f32_16x16x4_f32(
            false, a, false, bb, (short)0, acc, false, false);
      }
      nxt[t] = acc * 0.9f;             // per-step decay
      fin[t] += nxt[t];
    }
    __syncthreads();                   // everyone done reading ldsCur
#pragma unroll
    for (int t = 0; t < 8; ++t) {
      int col = t * 16 + colIn;
#pragma unroll
      for (int r = 0; r < 8; ++r)
        ldsCur[(rBase + rofs + r) * NF_ + col] = nxt[t][r];
    }
    __syncthreads();
    (void)decay;
  }

  float* outB = finalMask + (size_t)b * NF_ * NF_;
#pragma unroll
  for (int t = 0; t < 8; ++t) {
    int col = t * 16 + colIn;
#pragma unroll
    for (int r = 0; r < 8; ++r)
      outB[(rBase + rofs + r) * NF_ + col] = fin[t][r];
  }
}

// ---------------------------------------------------------------------------
// Kernel D: the 536 MB streaming kernel.
// Grid (4 m-tiles, 32 batches), 256 threads = 8 waves.
// Each workgroup owns (b, m in [mBase, mBase+32)) and loops over all n.
// Per (n,m): wave loads the 1 KB edge vector (2x b128 per lane), dots with
// the lane-resident query slice, butterfly-reduces across 32 lanes, applies
// sigmoid * final_mask, and lane (m-mBase) accumulates att_out[m,g].
// Cross-wave reduction via LDS ds_add_f32 atomics.
// ---------------------------------------------------------------------------
__global__ __launch_bounds__(256) void weit_att_kernel(
    const float* __restrict__ fe, const float* __restrict__ q,
    const float* __restrict__ fm, const float* __restrict__ attIn,
    float* __restrict__ attOut) {
  const int b     = blockIdx.y;
  const int mBase = blockIdx.x * 32;
  const int lane  = threadIdx.x & 31;
  const int wave  = threadIdx.x >> 5;

  // each lane owns query[b, lane*8 .. lane*8+7]
  const float4* qv = (const float4*)(q + (size_t)b * DE_);
  float4 q0 = qv[lane * 2];
  float4 q1 = qv[lane * 2 + 1];

  __shared__ float sAcc[32 * 2];
  if (threadIdx.x < 64) sAcc[threadIdx.x] = 0.0f;
  __syncthreads();

  float acc0 = 0.f, acc1 = 0.f;
  for (int n = wave; n < NF_; n += 8) {
    float ai0 = attIn[((size_t)b * NF_ + n) * G_ + 0];
    float ai1 = attIn[((size_t)b * NF_ + n) * G_ + 1];
    const float* base  = fe + (((size_t)b * NF_ + n) * NF_ + mBase) * DE_;
    const float* fmRow = fm + ((size_t)b * NF_ + n) * NF_ + mBase;
    for (int mi = 0; mi < 32; ++mi) {
      const float4* p = (const float4*)(base + (size_t)mi * DE_) + lane * 2;
      float4 e0 = p[0];
      float4 e1 = p[1];
      if (mi + 1 < 32)  // pull next 1 KB row toward L2/L0 while we reduce
        __builtin_prefetch(base + (size_t)(mi + 1) * DE_ + lane * 8, 0, 1);
      float s = e0.x * q0.x + e0.y * q0.y + e0.z * q0.z + e0.w * q0.w +
                e1.x * q1.x + e1.y * q1.y + e1.z * q1.z + e1.w * q1.w;
#pragma unroll
      for (int off = 16; off > 0; off >>= 1)
        s += __shfl_xor(s, off, 32);   // all 32 lanes end with the full dot
      float w = fmRow[mi] / (1.0f + __expf(-s));
      if (lane == mi) { acc0 += ai0 * w; acc1 += ai1 * w; }
    }
  }
  atomicAdd(&sAcc[lane * 2 + 0], acc0);   // ds_add_f32 across 8 waves
  atomicAdd(&sAcc[lane * 2 + 1], acc1);
  __syncthreads();
  if (threadIdx.x < 64) {
    int mi = threadIdx.x >> 1, g = threadIdx.x & 1;
    attOut[(((size_t)b * NF_) + mBase + mi) * G_ + g] = sAcc[threadIdx.x];
  }
}

// ---------------------------------------------------------------------------
// Kernel E: normalize by per-(b,g) row max (clamped >= 1), blend stack,
// copy stack_ptr through, zero the two dummy outputs.
// One block per batch; thread t -> (m = t>>1, g = t&1).
// ---------------------------------------------------------------------------
__global__ __launch_bounds__(256) void finalize_kernel(
    const float* __restrict__ attOut, const float* __restrict__ attStack,
    const float* __restrict__ stackPtr, float* __restrict__ outStack,
    float* __restrict__ outPtr, float* __restrict__ outZero1,
    float* __restrict__ outZero2) {
  int b = blockIdx.x;
  int t = threadIdx.x;
  __shared__ float sVal[NF_ * G_];
  __shared__ float sNorm[G_];
  __shared__ float sp[SL_];

  float v = attOut[(size_t)b * NF_ * G_ + t];
  sVal[t] = v;
  if (t < SL_) sp[t] = stackPtr[(size_t)b * SL_ + t];
  __syncthreads();
  if (t < G_) {
    float mx = -3.4e38f;
    for (int m = 0; m < NF_; ++m) mx = fmaxf(mx, sVal[m * G_ + t]);
    sNorm[t] = (mx <= 1.0f) ? 1.0f : mx;
  }
  __syncthreads();

  int g = t & 1;
  float a = v / sNorm[g];
  const float* stk = attStack + (size_t)b * NF_ * G_ * SL_ + (size_t)t * SL_;
  float* o = outStack + (size_t)b * NF_ * G_ * SL_ + (size_t)t * SL_;
#pragma unroll
  for (int s = 0; s < SL_; ++s) {
    float spv = sp[s];
    o[s] = a * spv + stk[s] * (1.0f - spv);
  }
  if (t < SL_) outPtr[(size_t)b * SL_ + t] = sp[t];
  outZero1[(size_t)b * DH_ + t] = 0.f;
  outZero1[(size_t)b * DH_ + 256 + t] = 0.f;
  outZero2[(size_t)b * DH_ + t] = 0.f;
  outZero2[(size_t)b * DH_ + 256 + t] = 0.f;
}

// ---------------------------------------------------------------------------
// Launch. Input order: vision_feat, feat, feat_edge, c_i, relation_mask,
// att_stack, stack_ptr, mem_in, map_c_w, map_c_b.
// Output: att_stack_new (65536) | stack_ptr (256) | zeros (16384) | zeros (16384).
// ---------------------------------------------------------------------------
extern "C" void kernel_launch(void* const* d_in, const int* in_sizes, int n_in,
                              void* d_out, int out_size, void* d_ws, size_t ws_size,
                              hipStream_t stream) {
  const float* feat_edge = (const float*)d_in[2];
  const float* c_i       = (const float*)d_in[3];
  const float* rel_mask  = (const float*)d_in[4];
  const float* att_stack = (const float*)d_in[5];
  const float* stack_ptr = (const float*)d_in[6];
  const float* map_c_w   = (const float*)d_in[8];
  const float* map_c_b   = (const float*)d_in[9];

  float* out = (float*)d_out;
  float* outStack = out;                               // 32*128*2*8
  float* outPtr   = out + (size_t)B_ * NF_ * G_ * SL_; // 32*8
  float* outZ1    = outPtr + (size_t)B_ * SL_;         // 32*512
  float* outZ2    = outZ1 + (size_t)B_ * DH_;          // 32*512

  float* ws      = (float*)d_ws;
  float* q       = ws;                                  // 8192
  float* attIn   = q + (size_t)B_ * DE_;                // 8192
  float* fmask   = attIn + (size_t)B_ * NF_ * G_;       // 524288
  float* attOut  = fmask + (size_t)B_ * NF_ * NF_;      // 8192

  query_kernel<<<(B_ * DE_) / 256, 256, 0, stream>>>(c_i, map_c_w, map_c_b, q);
  attin_kernel<<<(B_ * NF_ * G_) / 256, 256, 0, stream>>>(att_stack, stack_ptr, attIn);
  mask_power_kernel<<<B_, 256, 0, stream>>>(rel_mask, fmask);
  weit_att_kernel<<<dim3(NF_ / 32, B_), 256, 0, stream>>>(feat_edge, q, fmask, attIn, attOut);
  finalize_kernel<<<B_, 256, 0, stream>>>(attOut, att_stack, stack_ptr,
                                          outStack, outPtr, outZ1, outZ2);
}